// HalfWarper_283467842026
// MI455X (gfx1250) — compile-verified
//
#include <hip/hip_runtime.h>
#include <stdint.h>

// Problem constants (fixed by the reference: B=2, H=720, W=1280).
#define B_  2
#define H_  720
#define W_  1280
#define HW_ (H_ * W_)
#define EPS_ 1e-7f

// Morphology tiling: 16x16 output tile + 2-pixel halo (supports k<=5; the
// harness uses k=5). Larger k falls back to direct global-memory windows.
#define MT    16
#define HALO  2
#define LW_   (MT + 2 * HALO)   // 20

// ---------------------------------------------------------------------------
// CDNA5 async global->LDS copy (ASYNCcnt-tracked). One dword per lane lands
// directly in LDS at the byte offset taken from the low 32 bits of the
// generic pointer to the __shared__ element (addrspacecast LDS->flat keeps
// the LDS byte offset in addr[31:0]).
// ---------------------------------------------------------------------------
__device__ __forceinline__ void load_f32_to_lds(float* lptr, const float* g) {
#if defined(__gfx1250__)
    unsigned lds_off = (unsigned)(uintptr_t)lptr;
    asm volatile("global_load_async_to_lds_b32 %0, %1, off"
                 :: "v"(lds_off), "v"(g)
                 : "memory");
#else
    *lptr = *g;
#endif
}

__device__ __forceinline__ void wait_async_zero() {
#if defined(__gfx1250__)
    asm volatile("s_wait_asynccnt 0" ::: "memory");
#endif
}

// Fully unrolled 5x5 window min/max over the LDS tile (k==5 fast path).
// tx,ty are the top-left of the window inside the tile (i.e. threadIdx +
// HALO - 2 == threadIdx for HALO==2).
__device__ __forceinline__ float win5_min(const float* __restrict__ tile,
                                          int tx, int ty) {
    float m = __builtin_inff();
#pragma unroll
    for (int dy = 0; dy < 5; ++dy)
#pragma unroll
        for (int dx = 0; dx < 5; ++dx)
            m = fminf(m, tile[(ty + dy) * LW_ + (tx + dx)]);
    return m;
}
__device__ __forceinline__ float win5_max(const float* __restrict__ tile,
                                          int tx, int ty) {
    float m = -__builtin_inff();
#pragma unroll
    for (int dy = 0; dy < 5; ++dy)
#pragma unroll
        for (int dx = 0; dx < 5; ++dx)
            m = fmaxf(m, tile[(ty + dy) * LW_ + (tx + dx)]);
    return m;
}

// ---------------------------------------------------------------------------
// Kernel 1: zero the splat accumulators (8 planes of B*HW floats in d_ws).
// ---------------------------------------------------------------------------
__global__ __launch_bounds__(256) void k_zero(float4* __restrict__ p, int n4) {
    int i = blockIdx.x * blockDim.x + threadIdx.x;
    if (i < n4) p[i] = make_float4(0.f, 0.f, 0.f, 0.f);
}

// ---------------------------------------------------------------------------
// Kernel 2: fused softmax-splat for both warps.
//   acc layout per warp: [b][c][p], c = 0..2 img*w, c = 3 w.
// ---------------------------------------------------------------------------
__device__ __forceinline__ void splat4(float* __restrict__ acc,
                                       float fx, float fy,
                                       float r, float g, float b, float w) {
    float x0f = floorf(fx), y0f = floorf(fy);
    int   x0  = (int)x0f,   y0  = (int)y0f;
    float ax = fx - x0f, ay = fy - y0f;     // frac (exact)
    float bx = 1.0f - ax, by = 1.0f - ay;
    int   xs[2] = { x0, x0 + 1 };
    int   ys[2] = { y0, y0 + 1 };
    float wx[2] = { bx, ax };
    float wy[2] = { by, ay };
#pragma unroll
    for (int j = 0; j < 2; ++j) {
        if ((unsigned)ys[j] >= (unsigned)H_) continue;   // handles negatives too
#pragma unroll
        for (int i = 0; i < 2; ++i) {
            if ((unsigned)xs[i] >= (unsigned)W_) continue;
            float cw = wx[i] * wy[j];
            size_t o = (size_t)ys[j] * W_ + xs[i];
            atomicAdd(acc + (size_t)0 * HW_ + o, r * cw);
            atomicAdd(acc + (size_t)1 * HW_ + o, g * cw);
            atomicAdd(acc + (size_t)2 * HW_ + o, b * cw);
            atomicAdd(acc + (size_t)3 * HW_ + o, w * cw);
        }
    }
}

__global__ __launch_bounds__(256) void k_splat(
        const float* __restrict__ img0,  const float* __restrict__ img1,
        const float* __restrict__ flow0, const float* __restrict__ flow1,
        const float* __restrict__ z0,    const float* __restrict__ z1,
        float* __restrict__ acc01, float* __restrict__ acc10) {
    int tid = blockIdx.x * blockDim.x + threadIdx.x;
    if (tid >= B_ * HW_) return;
    int b = tid / HW_;
    int p = tid - b * HW_;
    int y = p / W_;
    int x = p - y * W_;

    // f01: splat img0 with flow1_ (channels swapped: x-disp=ch1, y-disp=ch0), weight exp(z1)
    {
        float fx = (float)x + 0.5f * flow1[(size_t)(b * 2 + 1) * HW_ + p];
        float fy = (float)y + 0.5f * flow1[(size_t)(b * 2 + 0) * HW_ + p];
        float w  = expf(z1[(size_t)b * HW_ + p]);
        float r  = img0[(size_t)(b * 3 + 0) * HW_ + p] * w;
        float g  = img0[(size_t)(b * 3 + 1) * HW_ + p] * w;
        float bl = img0[(size_t)(b * 3 + 2) * HW_ + p] * w;
        splat4(acc01 + (size_t)b * 4 * HW_, fx, fy, r, g, bl, w);
    }
    // f10: splat img1 with flow0_, weight exp(z0)
    {
        float fx = (float)x + 0.5f * flow0[(size_t)(b * 2 + 1) * HW_ + p];
        float fy = (float)y + 0.5f * flow0[(size_t)(b * 2 + 0) * HW_ + p];
        float w  = expf(z0[(size_t)b * HW_ + p]);
        float r  = img1[(size_t)(b * 3 + 0) * HW_ + p] * w;
        float g  = img1[(size_t)(b * 3 + 1) * HW_ + p] * w;
        float bl = img1[(size_t)(b * 3 + 2) * HW_ + p] * w;
        splat4(acc10 + (size_t)b * 4 * HW_, fx, fy, r, g, bl, w);
    }
}

// ---------------------------------------------------------------------------
// Kernel 3: normalize + write f01i (ch6-8), f10i (ch9-11), flow0_ (12-13),
// flow1_ (14-15), pre-masks (16,17). Output layout [B,18,H,W].
// ---------------------------------------------------------------------------
__global__ __launch_bounds__(256) void k_norm(
        const float* __restrict__ acc01, const float* __restrict__ acc10,
        const float* __restrict__ flow0, const float* __restrict__ flow1,
        float* __restrict__ out) {
    int tid = blockIdx.x * blockDim.x + threadIdx.x;
    if (tid >= B_ * HW_) return;
    int b = tid / HW_;
    int p = tid - b * HW_;
    const float* a0 = acc01 + (size_t)b * 4 * HW_;
    const float* a1 = acc10 + (size_t)b * 4 * HW_;
    float* ob = out + (size_t)b * 18 * HW_;

    float s0 = a0[(size_t)3 * HW_ + p];
    float s1 = a1[(size_t)3 * HW_ + p];
    float d0 = s0 + EPS_;
    float d1 = s1 + EPS_;
#pragma unroll
    for (int c = 0; c < 3; ++c) {
        ob[(size_t)(6 + c) * HW_ + p] = a0[(size_t)c * HW_ + p] / d0;
        ob[(size_t)(9 + c) * HW_ + p] = a1[(size_t)c * HW_ + p] / d1;
    }
    ob[(size_t)16 * HW_ + p] = s0 / d0;   // f01 pre-mask (opened later)
    ob[(size_t)17 * HW_ + p] = s1 / d1;   // f10 pre-mask
    ob[(size_t)12 * HW_ + p] = 0.5f * flow0[(size_t)(b * 2 + 0) * HW_ + p];
    ob[(size_t)13 * HW_ + p] = 0.5f * flow0[(size_t)(b * 2 + 1) * HW_ + p];
    ob[(size_t)14 * HW_ + p] = 0.5f * flow1[(size_t)(b * 2 + 0) * HW_ + p];
    ob[(size_t)15 * HW_ + p] = 0.5f * flow1[(size_t)(b * 2 + 1) * HW_ + p];
}

// ---------------------------------------------------------------------------
// Shared tile stage for the morphology kernels: async-load a (MT+4)x(MT+4)
// clamped-halo tile of `src` into LDS, wait ASYNCcnt==0, barrier.
// Clamp-to-edge is exactly equivalent to the reference's inf-padded SAME
// reduce_window for both min and max (duplicated edge values are already in
// every border window).
// ---------------------------------------------------------------------------
__device__ __forceinline__ void stage_tile(float* __restrict__ tile,
                                           const float* __restrict__ src) {
    int tx0 = blockIdx.x * MT - HALO, ty0 = blockIdx.y * MT - HALO;
    int tid = threadIdx.y * MT + threadIdx.x;
#pragma unroll
    for (int it = 0; it < 2; ++it) {           // 2*256 >= LW_*LW_ (400)
        int i = tid + it * (MT * MT);
        if (i < LW_ * LW_) {
            int ly = i / LW_, lx = i - ly * LW_;
            int sy = min(max(ty0 + ly, 0), H_ - 1);
            int sx = min(max(tx0 + lx, 0), W_ - 1);
            load_f32_to_lds(&tile[i], src + (size_t)sy * W_ + sx);
        }
    }
    wait_async_zero();
    __syncthreads();
}

// ---------------------------------------------------------------------------
// Kernel 4: erode (k x k min) of the two pre-masks (out ch16/17) into
// ebuf[pz*HW]. grid.z = B*2 : pz -> b = pz>>1, which = pz&1.
// ---------------------------------------------------------------------------
__global__ __launch_bounds__(256) void k_erode(
        const float* __restrict__ out, float* __restrict__ ebuf,
        const int* __restrict__ kp) {
    __shared__ float tile[LW_ * LW_];
    int k  = kp[0];
    int lo = (k > 0) ? (k - 1) / 2 : 0;
    int hi = (k > 0) ? k / 2 : 0;
    int pz = blockIdx.z, b = pz >> 1, which = pz & 1;
    const float* src = out + ((size_t)b * 18 + 16 + which) * HW_;
    float* dst = ebuf + (size_t)pz * HW_;
    int x = blockIdx.x * MT + threadIdx.x;
    int y = blockIdx.y * MT + threadIdx.y;

    if (k <= 2 * HALO + 1) {                       // LDS path (k <= 5)
        stage_tile(tile, src);
        if (x < W_ && y < H_) {
            float m;
            if (k == 5) {                          // unrolled fast path
                m = win5_min(tile, threadIdx.x, threadIdx.y);
            } else {
                m = __builtin_inff();
                for (int dy = -lo; dy <= hi; ++dy)
                    for (int dx = -lo; dx <= hi; ++dx)
                        m = fminf(m, tile[(threadIdx.y + HALO + dy) * LW_ +
                                          (threadIdx.x + HALO + dx)]);
            }
            dst[(size_t)y * W_ + x] = m;
        }
    } else if (x < W_ && y < H_) {                 // direct-global fallback
        float m = __builtin_inff();
        for (int dy = -lo; dy <= hi; ++dy) {
            int sy = y + dy;
            if ((unsigned)sy >= (unsigned)H_) continue;
            for (int dx = -lo; dx <= hi; ++dx) {
                int sx = x + dx;
                if ((unsigned)sx >= (unsigned)W_) continue;
                m = fminf(m, src[(size_t)sy * W_ + sx]);
            }
        }
        dst[(size_t)y * W_ + x] = m;
    }
}

// ---------------------------------------------------------------------------
// Kernel 5: dilate (k x k max) of eroded masks + final composition.
//   which==0: write f01m (ch16) and base0 (ch0-2)
//   which==1: write f10m (ch17) and base1 (ch3-5)
// ---------------------------------------------------------------------------
__global__ __launch_bounds__(256) void k_dilate_compose(
        const float* __restrict__ ebuf, float* __restrict__ out,
        const int* __restrict__ kp) {
    __shared__ float tile[LW_ * LW_];
    int k  = kp[0];
    int lo = (k > 0) ? (k - 1) / 2 : 0;
    int hi = (k > 0) ? k / 2 : 0;
    int pz = blockIdx.z, b = pz >> 1, which = pz & 1;
    const float* src = ebuf + (size_t)pz * HW_;
    int x = blockIdx.x * MT + threadIdx.x;
    int y = blockIdx.y * MT + threadIdx.y;
    float mval;
    bool active = (x < W_ && y < H_);

    if (k <= 2 * HALO + 1) {                       // LDS path (k <= 5)
        stage_tile(tile, src);
        if (!active) return;
        if (k == 5) {                              // unrolled fast path
            mval = win5_max(tile, threadIdx.x, threadIdx.y);
        } else {
            mval = -__builtin_inff();
            for (int dy = -lo; dy <= hi; ++dy)
                for (int dx = -lo; dx <= hi; ++dx)
                    mval = fmaxf(mval, tile[(threadIdx.y + HALO + dy) * LW_ +
                                            (threadIdx.x + HALO + dx)]);
        }
    } else {                                       // direct-global fallback
        if (!active) return;
        mval = -__builtin_inff();
        for (int dy = -lo; dy <= hi; ++dy) {
            int sy = y + dy;
            if ((unsigned)sy >= (unsigned)H_) continue;
            for (int dx = -lo; dx <= hi; ++dx) {
                int sx = x + dx;
                if ((unsigned)sx >= (unsigned)W_) continue;
                mval = fmaxf(mval, src[(size_t)sy * W_ + sx]);
            }
        }
    }

    size_t p = (size_t)y * W_ + x;
    float* ob = out + (size_t)b * 18 * HW_;
    const float* f01 = ob + (size_t)6 * HW_;
    const float* f10 = ob + (size_t)9 * HW_;
    if (which == 0) {
        ob[(size_t)16 * HW_ + p] = mval;
#pragma unroll
        for (int c = 0; c < 3; ++c)
            ob[(size_t)c * HW_ + p] =
                mval * f01[(size_t)c * HW_ + p] +
                (1.0f - mval) * f10[(size_t)c * HW_ + p];
    } else {
        ob[(size_t)17 * HW_ + p] = mval;
#pragma unroll
        for (int c = 0; c < 3; ++c)
            ob[(size_t)(3 + c) * HW_ + p] =
                mval * f10[(size_t)c * HW_ + p] +
                (1.0f - mval) * f01[(size_t)c * HW_ + p];
    }
}

// ---------------------------------------------------------------------------
// Host entry. d_in order: img0, img1, flow0, flow1, z0, z1, k.
// d_ws layout (floats): acc01 [B*4*HW] | acc10 [B*4*HW] | ebuf [B*2*HW]
//   = 10*B*HW floats = 73.7 MB.
// ---------------------------------------------------------------------------
extern "C" void kernel_launch(void* const* d_in, const int* in_sizes, int n_in,
                              void* d_out, int out_size, void* d_ws, size_t ws_size,
                              hipStream_t stream) {
    const float* img0  = (const float*)d_in[0];
    const float* img1  = (const float*)d_in[1];
    const float* flow0 = (const float*)d_in[2];
    const float* flow1 = (const float*)d_in[3];
    const float* z0    = (const float*)d_in[4];
    const float* z1    = (const float*)d_in[5];
    const int*   kp    = (const int*)d_in[6];
    float* out = (float*)d_out;
    float* ws  = (float*)d_ws;

    float* acc01 = ws;
    float* acc10 = ws + (size_t)4 * B_ * HW_;
    float* ebuf  = ws + (size_t)8 * B_ * HW_;

    int n4 = (8 * B_ * HW_) / 4;
    k_zero<<<(n4 + 255) / 256, 256, 0, stream>>>((float4*)ws, n4);

    int np = B_ * HW_;
    k_splat<<<(np + 255) / 256, 256, 0, stream>>>(img0, img1, flow0, flow1,
                                                  z0, z1, acc01, acc10);
    k_norm<<<(np + 255) / 256, 256, 0, stream>>>(acc01, acc10, flow0, flow1, out);

    dim3 blk(MT, MT, 1);
    dim3 grd((W_ + MT - 1) / MT, (H_ + MT - 1) / MT, B_ * 2);
    k_erode<<<grd, blk, 0, stream>>>(out, ebuf, kp);
    k_dilate_compose<<<grd, blk, 0, stream>>>(ebuf, out, kp);
}